// CrfHead_85822036509475
// MI455X (gfx1250) — compile-verified
//
#include <hip/hip_runtime.h>

typedef float        f32x4 __attribute__((ext_vector_type(4)));
typedef unsigned int u32x4 __attribute__((ext_vector_type(4)));
typedef int          i32x4 __attribute__((ext_vector_type(4)));
typedef int          i32x8 __attribute__((ext_vector_type(8)));

#define TAGS 256
#define ROWS (128 * 1024)

__global__ void CrfHead_85822036509475_kernel(const float* __restrict__ launch,
                                              const float* __restrict__ trans,
                                              float* __restrict__ out) {
    // Dynamic LDS: 256x256 f32 transitions table (256 KB), at LDS offset 0.
    extern __shared__ float s_trans[];

    // ---- TDM: DMA the full transitions matrix (256 KB) into LDS ----------
    // One wave (wave 0) issues a single tensor_load_to_lds; TDM ignores EXEC
    // and performs the full descriptor. Other waves skip via execz branch.
    if (threadIdx.x == 0) {
        unsigned long long ga = (unsigned long long)(const void*)trans;

        // D# group 0 (128b): count=1 | lds_addr=0 | global_addr | type=2
        u32x4 g0;
        g0[0] = 1u;                                    // [1:0] count=1, rest 0
        g0[1] = 0u;                                    // [63:32] lds_addr = 0
        g0[2] = (unsigned int)(ga & 0xFFFFFFFFu);      // [95:64] addr[31:0]
        g0[3] = (unsigned int)((ga >> 32) & 0x01FFFFFFu) // [120:96] addr[56:32]
              | (2u << 30);                            // [127:126] type=2 (image)

        // D# group 1 (256b): 2D tensor 256x256, tile 256x256, elem=4B
        i32x8 g1;
        g1[0] = 0x00020000;          // workgroup_mask=0, data_size=2 (4B)
        g1[1] = (int)(256u << 16);   // tensor_dim0 = 256 (bits [63:48])
        g1[2] = (int)(256u << 16);   // tensor_dim0 hi=0; tensor_dim1=256 ([95:80])
        g1[3] = (int)(256u << 16);   // tensor_dim1 hi=0; tile_dim0=256 ([127:112])
        g1[4] = 256;                 // tile_dim1 = 256 ([143:128]); tile_dim2=0
        g1[5] = 256;                 // tensor_dim0_stride = 256 ([191:160])
        g1[6] = 0;                   // stride0 hi=0; tensor_dim1_stride=0 (2D)
        g1[7] = 0;

        i32x4 z4 = {0, 0, 0, 0};
#if __clang_major__ >= 23
        i32x8 z8 = {0, 0, 0, 0, 0, 0, 0, 0};
        __builtin_amdgcn_tensor_load_to_lds(g0, g1, z4, z4, z8, 0);
#else
        __builtin_amdgcn_tensor_load_to_lds(g0, g1, z4, z4, 0);
#endif
        __builtin_amdgcn_s_wait_tensorcnt(0);
    }
    __syncthreads();

    // ---- Streaming phase: one wave32 per row of 256 floats ----------------
    const int lane  = threadIdx.x & 31;
    const int wid   = threadIdx.x >> 5;
    const int gwave = blockIdx.x * (blockDim.x >> 5) + wid;
    const int nwave = gridDim.x * (blockDim.x >> 5);

    for (int row = gwave; row < ROWS; row += nwave) {
        const f32x4* src = (const f32x4*)(launch + (size_t)row * TAGS);
        // lane l covers elements [4l..4l+3] and [128+4l..128+4l+3]
        f32x4 a = __builtin_nontemporal_load(src + lane);
        f32x4 b = __builtin_nontemporal_load(src + 32 + lane);

        // local argmax over this lane's 8 elements (first occurrence wins)
        float mv = a[0];
        int   mi = lane * 4;
#pragma unroll
        for (int j = 1; j < 4; ++j)
            if (a[j] > mv) { mv = a[j]; mi = lane * 4 + j; }
#pragma unroll
        for (int j = 0; j < 4; ++j)
            if (b[j] > mv) { mv = b[j]; mi = 128 + lane * 4 + j; }

        // wave32 butterfly max-reduce; tie-break to the lower index so the
        // result matches jnp.argmax first-occurrence semantics
#pragma unroll
        for (int off = 16; off > 0; off >>= 1) {
            float ov = __shfl_xor(mv, off, 32);
            int   oi = __shfl_xor(mi, off, 32);
            if (ov > mv || (ov == mv && oi < mi)) { mv = ov; mi = oi; }
        }

        // gather transitions[mi, :] from LDS (ds_load_b128 x2 per lane)
        const f32x4* t = (const f32x4*)(s_trans + (size_t)mi * TAGS);
        f32x4 ta = t[lane];
        f32x4 tb = t[32 + lane];

        f32x4* dst = (f32x4*)(out + (size_t)row * TAGS);
        __builtin_nontemporal_store(a + ta, dst + lane);
        __builtin_nontemporal_store(b + tb, dst + 32 + lane);
    }
}

extern "C" void kernel_launch(void* const* d_in, const int* in_sizes, int n_in,
                              void* d_out, int out_size, void* d_ws, size_t ws_size,
                              hipStream_t stream) {
    const float* launch = (const float*)d_in[0];  // [128,1024,256] f32
    const float* trans  = (const float*)d_in[1];  // [256,256] f32
    float*       out    = (float*)d_out;          // [128,1024,256] f32
    (void)in_sizes; (void)n_in; (void)out_size; (void)d_ws; (void)ws_size;

    const size_t shmem = (size_t)TAGS * TAGS * sizeof(float);  // 256 KB LDS
    hipFuncSetAttribute((const void*)CrfHead_85822036509475_kernel,
                        hipFuncAttributeMaxDynamicSharedMemorySize, (int)shmem);

    // 512 blocks x 1024 threads = 16384 waves; 8 rows per wave (grid-stride).
    CrfHead_85822036509475_kernel<<<512, 1024, shmem, stream>>>(launch, trans, out);
}